// SparseTechnicalNeuron_28441273434821
// MI455X (gfx1250) — compile-verified
//
#include <hip/hip_runtime.h>
#include <cstdint>

typedef __attribute__((ext_vector_type(2))) float v2f;
typedef __attribute__((ext_vector_type(8))) float v8f;

#define B_ROWS 40000
#define N_COLS 2500
#define KCON   50
#define KPAD   52   // 13 chunks of 4, zero-padded

// One wave (32 lanes) computes 16 rows' dot products via v_wmma_f32_16x16x4_f32.
// A (16x4): lane%16 = row M, VGPR0/1 = K=k0/k0+1 (lanes 0-15), K=k0+2/k0+3 (lanes 16-31).
// B (4x16): weight w[k] broadcast across all N columns -> every D column = dot product.
// D (16x16 f32): VGPR r, lane 0 -> M=r ; lane 16 -> M=8+r. We read column N=0.
__global__ __launch_bounds__(256) void SparseTechnicalNeuron_wmma_kernel(
    const float* __restrict__ act,     // [B_ROWS, N_COLS]
    const int*   __restrict__ idx,     // [KCON]
    const float* __restrict__ w,       // [KCON]
    const float* __restrict__ sens,    // [1]
    const float* __restrict__ thr,     // [1]
    float*       __restrict__ out)     // [B_ROWS]
{
    __shared__ int   s_idx[KPAD];
    __shared__ float s_w[KPAD];

    const int tid = threadIdx.x;
    if (tid < KPAD) {
        if (tid < KCON) { s_idx[tid] = idx[tid]; s_w[tid] = w[tid]; }
        else            { s_idx[tid] = 0;        s_w[tid] = 0.0f;   }
    }
    __syncthreads();

    const int lane = tid & 31;
    const int wave = tid >> 5;        // 8 waves per 256-thread block (wave32)
    const int half = lane >> 4;       // 0 for lanes 0-15, 1 for lanes 16-31
    const int mrow = lane & 15;       // row within the 16-row tile

    const int base_row = (blockIdx.x * 8 + wave) * 16;
    int row = base_row + mrow;
    int rclamp = (row < B_ROWS) ? row : (B_ROWS - 1);   // keep EXEC full for WMMA
    const float* __restrict__ arow = act + (size_t)rclamp * N_COLS;

    v8f acc = {0.f, 0.f, 0.f, 0.f, 0.f, 0.f, 0.f, 0.f};

    #pragma unroll
    for (int c = 0; c < 13; ++c) {
        const int k0 = c * 4 + half * 2;
        v2f a;
        a.x = arow[s_idx[k0]];
        a.y = arow[s_idx[k0 + 1]];
        v2f bm;
        bm.x = s_w[k0];
        bm.y = s_w[k0 + 1];
        // 8 args: (neg_a, A, neg_b, B, c_mod, C, reuse_a, reuse_b)
        acc = __builtin_amdgcn_wmma_f32_16x16x4_f32(
            false, a, false, bm, (short)0, acc, false, false);
    }

    const float s = sens[0];
    const float t = thr[0];

    // Column N=0 lives in lanes 0 (rows base..base+7) and 16 (rows base+8..base+15).
    if (mrow == 0) {
        const int m0 = base_row + half * 8;
        #pragma unroll
        for (int r = 0; r < 8; ++r) {
            const int m = m0 + r;
            if (m < B_ROWS) {
                const float z = acc[r] * s - t;
                out[m] = 1.0f / (1.0f + __expf(-z));
            }
        }
    }
}

extern "C" void kernel_launch(void* const* d_in, const int* in_sizes, int n_in,
                              void* d_out, int out_size, void* d_ws, size_t ws_size,
                              hipStream_t stream) {
    // setup_inputs() order:
    //   0: x [B,F] f32 (UNUSED by the reference math)
    //   1: all_activations [B,N] f32
    //   2: connection_indices [K] int
    //   3: connection_weights [K] f32
    //   4: sensitivity [1] f32
    //   5: threshold [1] f32
    const float* act  = (const float*)d_in[1];
    const int*   idx  = (const int*)  d_in[2];
    const float* w    = (const float*)d_in[3];
    const float* sens = (const float*)d_in[4];
    const float* thr  = (const float*)d_in[5];
    float* out = (float*)d_out;

    // 8 waves/block * 16 rows/wave = 128 rows per block
    const int rows_per_block = 128;
    const int grid = (B_ROWS + rows_per_block - 1) / rows_per_block;  // 313
    SparseTechnicalNeuron_wmma_kernel<<<grid, 256, 0, stream>>>(act, idx, w, sens, thr, out);
}